// SphericalHarmonicEncoder_16716012716489
// MI455X (gfx1250) — compile-verified
//
#include <hip/hip_runtime.h>
#include <cstdint>

// ---------------------------------------------------------------------------
// Spherical-harmonic + radial positional encoder for MI455X (gfx1250).
// One lane per point; results staged in LDS and drained to HBM with
// CDNA5 async LDS->global b128 stores (512B coalesced bursts / instruction).
// Radial sincos uses FMA Cody-Waite reduction + native v_sin/v_cos instead
// of OCML's Payne-Hanek path (args bounded by ~1.1e6 rad -> k < 2^24).
// ---------------------------------------------------------------------------

#define LMAX 12            // legendre order (L and M)
#define NCH  256           // total channels
#define NSH  (LMAX*LMAX)   // 144 SH channels
#define NRAD (NCH - NSH)   // 112 radial channels
#define NHALF (NRAD/2)     // 56 sin + 56 cos

// Padded LDS row: 256 payload + 4 pad floats -> 1040B stride (16B aligned,
// stride mod 64 banks = 4 -> <=2-way conflict on the staging stores).
#define ROWF 260
#define ROWB (ROWF*4)

// ---- compile-time K_{l,m} table (sqrt((2l+1)/(4pi) (l-m)!/(l+m)!), x sqrt2 for m>0)
namespace shk {
constexpr double PI_ = 3.14159265358979323846;
constexpr double cfact(int n) { double r = 1.0; for (int i = 2; i <= n; ++i) r *= (double)i; return r; }
constexpr double csqrt(double x) {
    double r = x > 1.0 ? x : 1.0;
    for (int i = 0; i < 64; ++i) r = 0.5 * (r + x / r);
    return r;
}
struct KTab { float v[LMAX][LMAX]; };
constexpr KTab makeK() {
    KTab t{};
    for (int l = 0; l < LMAX; ++l)
        for (int m = 0; m <= l; ++m) {
            double k = csqrt((2.0 * l + 1.0) / (4.0 * PI_) * cfact(l - m) / cfact(l + m));
            if (m > 0) k *= 1.41421356237309504880;  // sqrt(2)
            t.v[l][m] = (float)k;
        }
    return t;
}
constexpr KTab KT = makeK();
}  // namespace shk

// Cody-Waite 2-term reduction (exact fma products; k < 2^24) + native trig.
__device__ __forceinline__ void sincos_big(float a, float* sn, float* cs) {
    const float INV2PI = 0.15915493667125702f;   // fl(1/2pi)
    const float C1     = 6.2831854820251465f;    // fl(2pi)
    const float C2     = -1.7484556000744487e-7f;// fl(2pi - (double)C1)
    float k = rintf(a * INV2PI);
    float x = fmaf(-k, C1, a);
    x = fmaf(-k, C2, x);                         // x in ~[-pi-0.04, pi+0.04]
    *sn = __sinf(x);                             // native v_sin_f32 path
    *cs = __cosf(x);                             // native v_cos_f32 path
}

__global__ __launch_bounds__(32)
void she_encode_kernel(const float* __restrict__ phi_in,
                       const float* __restrict__ theta_in,
                       const float* __restrict__ r_in,
                       float* __restrict__ out, int n) {
    __shared__ __align__(16) float stage[32 * ROWF];

    const int lane     = threadIdx.x;             // 0..31 (wave32)
    const int tileBase = blockIdx.x * 32;
    int p = tileBase + lane;
    if (p >= n) p = n - 1;                        // clamp: keep EXEC full, dup compute

    const float phi   = phi_in[p];
    const float theta = theta_in[p];
    const float rr    = r_in[p];

    // theta in [0,pi], phi in [-pi,pi]: native trig needs no reduction.
    const float st = __sinf(theta), ct = __cosf(theta);
    const float sp = __sinf(phi),   cp = __cosf(phi);

    float* row = &stage[lane * ROWF];

    // ---------------- spherical harmonics (m-major, all-FMA recurrences) ----
    float c_m = 1.0f, s_m = 0.0f;                 // cos(m phi), sin(m phi)
    float pmm = 1.0f;                             // P(m,m) chain (Condon-Shortley)

    auto emit = [&](int l, int m, float P) {
        const int base = l * (l + 1);
        if (m == 0) {
            row[base] = shk::KT.v[l][0] * P;
        } else {
            const float kp = shk::KT.v[l][m] * P; // sqrt2 folded into table
            row[base + m] = kp * c_m;             // m > 0 channel
            row[base - m] = kp * s_m;             // m < 0 channel
        }
    };

#pragma unroll
    for (int m = 0; m < LMAX; ++m) {
        float p0 = pmm;                           // P(m,m)
        emit(m, m, p0);
        if (m + 1 < LMAX) {
            float p1 = (float)(2 * m + 1) * ct * p0;  // P(m+1,m)
            emit(m + 1, m, p1);
#pragma unroll
            for (int l = m + 2; l < LMAX; ++l) {
                float pl = ((float)(2 * l - 1) * ct * p1 - (float)(l + m - 1) * p0)
                           * (1.0f / (float)(l - m));
                emit(l, m, pl);
                p0 = p1; p1 = pl;
            }
        }
        pmm = -(float)(2 * m + 1) * st * pmm;     // -> P(m+1,m+1)
        const float nc = c_m * cp - s_m * sp;     // angle addition
        s_m = s_m * cp + c_m * sp;
        c_m = nc;
    }

    // ------- radial encoding: omega_i = 1000 * 100^(-i/56) = 1000*2^(E*i) ---
    // Outer loop kept rolled (unroll 1) so omega stays a runtime exp2 (1 trans
    // op) instead of 56 pre-materialized literals blowing up VGPR pressure;
    // inner x4 unroll gives trans-pipe ILP and lets the 4 consecutive
    // row[] stores merge into ds_store_b128.
    {
        const float E = -0.1186402893f;           // -log2(100)/56
#pragma unroll 1
        for (int i0 = 0; i0 < NHALF; i0 += 4) {
#pragma unroll
            for (int j = 0; j < 4; ++j) {
                const int i = i0 + j;
                const float w = 1000.0f * exp2f(E * (float)i);  // native v_exp_f32
                float sn, cs;
                sincos_big(rr * w, &sn, &cs);
                row[NSH + i]         = sn;
                row[NSH + NHALF + i] = cs;
            }
        }
    }

    // ---------------- drain tile with CDNA5 async LDS->global stores --------
    // Wait for this wave's ds_stores to land in LDS before the async engine reads.
    asm volatile("s_wait_dscnt 0x0" ::: "memory");

    int nvalid = n - tileBase; if (nvalid > 32) nvalid = 32;

    // LDS flat pointer truncation -> LDS byte offset (aperture keeps it in [31:0]).
    unsigned lo   = (unsigned)(uintptr_t)(&stage[0]) + (unsigned)lane * 16u;
    uint64_t gadr = (uint64_t)(uintptr_t)(out + (size_t)tileBase * NCH) + (uint64_t)lane * 16u;

    for (int pp = 0; pp < nvalid; ++pp) {
        // each instruction: 32 lanes x 16B = 512B coalesced burst; offset: is
        // added to BOTH the LDS and the global address per the ISA pseudocode.
        asm volatile("global_store_async_from_lds_b128 %0, %1, off"
                     :: "v"(gadr), "v"(lo) : "memory");
        asm volatile("global_store_async_from_lds_b128 %0, %1, off offset:512"
                     :: "v"(gadr), "v"(lo) : "memory");
        gadr += (uint64_t)NCH * 4u;               // 1024B per point in global
        lo   += ROWB;                             // 1040B padded row in LDS
    }

    // Drain before LDS dealloc (S_ENDPGM also implies wait-idle).
    asm volatile("s_wait_asynccnt 0x0" ::: "memory");
}

extern "C" void kernel_launch(void* const* d_in, const int* in_sizes, int n_in,
                              void* d_out, int out_size, void* d_ws, size_t ws_size,
                              hipStream_t stream) {
    (void)n_in; (void)out_size; (void)d_ws; (void)ws_size;
    const float* phi   = (const float*)d_in[0];
    const float* theta = (const float*)d_in[1];
    const float* r     = (const float*)d_in[2];
    float* out = (float*)d_out;
    const int n = in_sizes[0];                    // B*N points (320000)
    const int blocks = (n + 31) / 32;
    she_encode_kernel<<<blocks, 32, 0, stream>>>(phi, theta, r, out, n);
}